// HeteroGraphSAGE_58763742544007
// MI455X (gfx1250) — compile-verified
//
#include <hip/hip_runtime.h>

#define N_APP 200000
#define N_DEV 100000
#define N_EML 100000
#define NEDGE 1000000
#define DD 64

typedef __attribute__((ext_vector_type(2))) float v2f;
typedef __attribute__((ext_vector_type(8))) float v8f;

// ---------------- edge-phase kernels (bandwidth bound) ----------------

__global__ void count_kernel(const int* __restrict__ dst, float* __restrict__ cnt, int E) {
    int e = blockIdx.x * blockDim.x + threadIdx.x;
    if (e < E) atomicAdd(&cnt[dst[e]], 1.0f);
}

// one wave (32 lanes) per edge; each lane moves a float2 (lanes cover 64 feats)
__global__ void scatter_kernel(const float* __restrict__ hsrc, const int* __restrict__ src,
                               const int* __restrict__ dst, float* __restrict__ sum, int E) {
    int t = blockIdx.x * blockDim.x + threadIdx.x;
    int e = t >> 5;
    int lane = t & 31;
    if (e >= E) return;
    int s = src[e];
    int d = dst[e];
    float2 v = ((const float2*)(hsrc + (size_t)s * DD))[lane];
    float* dp = sum + (size_t)d * DD + lane * 2;
    atomicAdd(dp, v.x);
    atomicAdd(dp + 1, v.y);
}

// ---------------- fused SAGE GEMM: out = [prev +] mean@Wn + self@Ws + b ----------------
// block: 128 threads = 4 waves; each block does 16 rows x 64 cols, wave w does cols [16w,16w+16)
// f32 WMMA 16x16x4, K=64 -> 16 k-steps x 2 GEMMs = 32 v_wmma per wave

__global__ __launch_bounds__(128)
void sage_gemm_kernel(const float* __restrict__ sum, const float* __restrict__ cnt,
                      const float* __restrict__ hself,
                      const float* __restrict__ Wn, const float* __restrict__ Ws,
                      const float* __restrict__ bias,
                      const float* __restrict__ prev, float* __restrict__ out,
                      int relu) {
    __shared__ float sM[16][66];   // mean-aggregated neighbor rows (padded: bank-conflict free)
    __shared__ float sS[16][66];   // self rows
    const int row0 = blockIdx.x * 16;
    const int tid  = threadIdx.x;

    // stage A operands: 16x64 each, 128 threads x 8 elems (uniform control flow -> EXEC all 1s)
    for (int i = tid; i < 16 * 64; i += 128) {
        int m = i >> 6, k = i & 63;
        float inv = 1.0f / fmaxf(cnt[row0 + m], 1.0f);
        sM[m][k] = sum[(row0 + m) * 64 + k] * inv;
        sS[m][k] = hself[(row0 + m) * 64 + k];
    }
    __syncthreads();

    const int wave = tid >> 5;      // 0..3 -> output column slab
    const int lane = tid & 31;
    const int hh   = lane >> 4;     // lane half
    const int mn   = lane & 15;     // M for A-frag, N for B/C-frag
    const int ncol = wave * 16 + mn;

    v8f acc = {};
#pragma unroll
    for (int k0 = 0; k0 < 64; k0 += 4) {
        const int ka = k0 + 2 * hh;         // ISA layout: VGPR j, half h holds K = j + 2h
        v2f aM, aS, bN, bS;
        aM.x = sM[mn][ka];      aM.y = sM[mn][ka + 1];
        aS.x = sS[mn][ka];      aS.y = sS[mn][ka + 1];
        bN.x = Wn[ka * 64 + ncol];  bN.y = Wn[(ka + 1) * 64 + ncol];
        bS.x = Ws[ka * 64 + ncol];  bS.y = Ws[(ka + 1) * 64 + ncol];
        acc = __builtin_amdgcn_wmma_f32_16x16x4_f32(false, aM, false, bN, (short)0, acc, false, false);
        acc = __builtin_amdgcn_wmma_f32_16x16x4_f32(false, aS, false, bS, (short)0, acc, false, false);
    }

    const float bn = bias[ncol];
#pragma unroll
    for (int r = 0; r < 8; ++r) {          // C/D layout: M = r + 8*half, N = ncol
        int row = row0 + r + 8 * hh;
        float v = acc[r] + bn;
        if (prev) v += prev[row * 64 + ncol];
        if (relu) v = fmaxf(v, 0.0f);
        out[row * 64 + ncol] = v;
    }
}

// ---------------- classifier: [N,64] @ [64,2] + bc ----------------

__global__ void classify_kernel(const float* __restrict__ h, const float* __restrict__ Wc,
                                const float* __restrict__ bc, float* __restrict__ out, int n) {
    int r = blockIdx.x * blockDim.x + threadIdx.x;
    if (r >= n) return;
    float a0 = bc[0], a1 = bc[1];
    const float* hr = h + (size_t)r * 64;
#pragma unroll
    for (int k = 0; k < 64; ++k) {
        float v = hr[k];
        a0 += v * Wc[2 * k];
        a1 += v * Wc[2 * k + 1];
    }
    out[2 * r]     = a0;
    out[2 * r + 1] = a1;
}

// ---------------- launch ----------------

extern "C" void kernel_launch(void* const* d_in, const int* in_sizes, int n_in,
                              void* d_out, int out_size, void* d_ws, size_t ws_size,
                              hipStream_t stream) {
    const float* emb_app = (const float*)d_in[0];
    const float* emb_dev = (const float*)d_in[1];
    const float* emb_eml = (const float*)d_in[2];
    const float* W1s = (const float*)d_in[3];   // [4,64,64]
    const float* W1n = (const float*)d_in[4];
    const float* b1  = (const float*)d_in[5];   // [4,64]
    const float* W2s = (const float*)d_in[6];
    const float* W2n = (const float*)d_in[7];
    const float* b2  = (const float*)d_in[8];
    const float* Wc  = (const float*)d_in[9];   // [64,2]
    const float* bc  = (const float*)d_in[10];  // [2]
    const int* e_uses_src   = (const int*)d_in[11];
    const int* e_uses_dst   = (const int*)d_in[12];
    const int* e_usedby_src = (const int*)d_in[13];
    const int* e_usedby_dst = (const int*)d_in[14];
    const int* e_has_src    = (const int*)d_in[15];
    const int* e_has_dst    = (const int*)d_in[16];
    const int* e_bel_src    = (const int*)d_in[17];
    const int* e_bel_dst    = (const int*)d_in[18];
    float* outp = (float*)d_out;

    // workspace layout
    float* ws = (float*)d_ws;
    float* cnt_dev  = ws;                                  // N_DEV
    float* cnt_appU = cnt_dev  + N_DEV;                    // N_APP
    float* cnt_eml  = cnt_appU + N_APP;                    // N_EML
    float* cnt_appB = cnt_eml  + N_EML;                    // N_APP
    float* sumA     = cnt_appB + N_APP;                    // N_DEV*64 (reused for dev & eml)
    float* sumB     = sumA + (size_t)N_DEV * DD;           // N_APP*64 (reused for app relations)
    float* h1_app   = sumB + (size_t)N_APP * DD;
    float* h1_dev   = h1_app + (size_t)N_APP * DD;
    float* h1_eml   = h1_dev + (size_t)N_DEV * DD;
    float* h2_app   = h1_eml + (size_t)N_EML * DD;

    const int CB = 256;
    const int cgrid = (NEDGE + CB - 1) / CB;
    const int sgrid = (NEDGE * 32) / CB;                   // 1 wave per edge
    const int M4096 = 64 * 64;

    // --- degree counts (identical for both layers) ---
    hipMemsetAsync(cnt_dev, 0, sizeof(float) * (size_t)(N_DEV + N_APP + N_EML + N_APP), stream);
    count_kernel<<<cgrid, CB, 0, stream>>>(e_uses_dst,   cnt_dev,  NEDGE);
    count_kernel<<<cgrid, CB, 0, stream>>>(e_usedby_dst, cnt_appU, NEDGE);
    count_kernel<<<cgrid, CB, 0, stream>>>(e_has_dst,    cnt_eml,  NEDGE);
    count_kernel<<<cgrid, CB, 0, stream>>>(e_bel_dst,    cnt_appB, NEDGE);

    // --- layer 1, rel 0: uses (app -> dev), ReLU ---
    hipMemsetAsync(sumA, 0, sizeof(float) * (size_t)N_DEV * DD, stream);
    scatter_kernel<<<sgrid, CB, 0, stream>>>(emb_app, e_uses_src, e_uses_dst, sumA, NEDGE);
    sage_gemm_kernel<<<N_DEV / 16, 128, 0, stream>>>(sumA, cnt_dev, emb_dev,
        W1n + 0 * M4096, W1s + 0 * M4096, b1 + 0 * 64, nullptr, h1_dev, 1);

    // --- layer 1, rel 1: used_by (dev -> app), accumulate later ---
    hipMemsetAsync(sumB, 0, sizeof(float) * (size_t)N_APP * DD, stream);
    scatter_kernel<<<sgrid, CB, 0, stream>>>(emb_dev, e_usedby_src, e_usedby_dst, sumB, NEDGE);
    sage_gemm_kernel<<<N_APP / 16, 128, 0, stream>>>(sumB, cnt_appU, emb_app,
        W1n + 1 * M4096, W1s + 1 * M4096, b1 + 1 * 64, nullptr, h1_app, 0);

    // --- layer 1, rel 3: email_of (eml -> app), add + ReLU ---
    hipMemsetAsync(sumB, 0, sizeof(float) * (size_t)N_APP * DD, stream);
    scatter_kernel<<<sgrid, CB, 0, stream>>>(emb_eml, e_bel_src, e_bel_dst, sumB, NEDGE);
    sage_gemm_kernel<<<N_APP / 16, 128, 0, stream>>>(sumB, cnt_appB, emb_app,
        W1n + 3 * M4096, W1s + 3 * M4096, b1 + 3 * 64, h1_app, h1_app, 1);

    // --- layer 1, rel 2: has_email (app -> eml), ReLU ---
    hipMemsetAsync(sumA, 0, sizeof(float) * (size_t)N_EML * DD, stream);
    scatter_kernel<<<sgrid, CB, 0, stream>>>(emb_app, e_has_src, e_has_dst, sumA, NEDGE);
    sage_gemm_kernel<<<N_EML / 16, 128, 0, stream>>>(sumA, cnt_eml, emb_eml,
        W1n + 2 * M4096, W1s + 2 * M4096, b1 + 2 * 64, nullptr, h1_eml, 1);

    // --- layer 2: only h2_app feeds the classifier (dev/eml outputs are dead) ---
    hipMemsetAsync(sumB, 0, sizeof(float) * (size_t)N_APP * DD, stream);
    scatter_kernel<<<sgrid, CB, 0, stream>>>(h1_dev, e_usedby_src, e_usedby_dst, sumB, NEDGE);
    sage_gemm_kernel<<<N_APP / 16, 128, 0, stream>>>(sumB, cnt_appU, h1_app,
        W2n + 1 * M4096, W2s + 1 * M4096, b2 + 1 * 64, nullptr, h2_app, 0);

    hipMemsetAsync(sumB, 0, sizeof(float) * (size_t)N_APP * DD, stream);
    scatter_kernel<<<sgrid, CB, 0, stream>>>(h1_eml, e_bel_src, e_bel_dst, sumB, NEDGE);
    sage_gemm_kernel<<<N_APP / 16, 128, 0, stream>>>(sumB, cnt_appB, h1_app,
        W2n + 3 * M4096, W2s + 3 * M4096, b2 + 3 * 64, h2_app, h2_app, 0);

    // --- classifier ---
    classify_kernel<<<(N_APP + CB - 1) / CB, CB, 0, stream>>>(h2_app, Wc, bc, outp, N_APP);
}